// TemporalDifferentModuleMSDeformAttnUnified_34196529610805
// MI455X (gfx1250) — compile-verified
//
#include <hip/hip_runtime.h>
#include <hip/hip_bf16.h>
#include <math.h>

// ---------------------------------------------------------------------------
// Temporal MS-Deformable-Attention for MI455X (gfx1250), fp32 end-to-end.
// GEMMs use V_WMMA_F32_16X16X4_F32 (exact fp32 -> matches fp32 reference),
// register-blocked 16x64 per wave (A fragment reused across 4 N-tiles).
// Sampling core: 1 wave32 per (query, head); lane = channel; coalesced gathers.
// ---------------------------------------------------------------------------

typedef __attribute__((ext_vector_type(2))) float v2f;
typedef __attribute__((ext_vector_type(8))) float v8f;

#define LQ_   12240
#define C_    256
#define HEADS_ 8
#define LEVELS_ 4
#define FRAMES_ 4
#define POINTS_ 4
#define DHEAD_ 32

// ---------------------------------------------------------------------------
// Generic fp32 GEMM: Y[M,N] = X[M,K] @ W[K,N] + bias[N]
// One wave per 16x64 output block (four 16x16 WMMA tiles sharing one A frag).
// K stepped by 4 via wmma_f32_16x16x4_f32. Requires N % 64 == 0, M % 16 == 0.
// A frag (16x4): lanes 0-15 hold row M=lane, K={k,k+1}; lanes 16-31 K={k+2,k+3}.
// B frag (4x16): lanes 0-15 hold col N=lane, K={k,k+1}; lanes 16-31 K={k+2,k+3}.
// C/D (16x16 f32): elem i -> row = m0 + i + 8*(lane>=16), col = n0 + (lane&15).
// ---------------------------------------------------------------------------
__global__ void gemm_f32_wmma(const float* __restrict__ X,
                              const float* __restrict__ W,
                              const float* __restrict__ bias,
                              float* __restrict__ Y,
                              int M, int N, int K) {
  const int wavesPerBlock = blockDim.x >> 5;
  const int wave = blockIdx.x * wavesPerBlock + (threadIdx.x >> 5);
  const int lane = threadIdx.x & 31;
  const int groupsN = N >> 6;                  // 64-wide column groups
  const int totalGroups = (M >> 4) * groupsN;
  if (wave >= totalGroups) return;             // wave-uniform: EXEC all-ones

  const int tm = wave / groupsN;
  const int tg = wave - tm * groupsN;
  const int m0 = tm << 4;
  const int n0 = tg << 6;
  const int half = lane >> 4;                  // 0: K pair {0,1}; 1: {2,3}
  const int l    = lane & 15;

  const float* __restrict__ xrow  = X + (size_t)(m0 + l) * K + (half ? 2 : 0);
  const float* __restrict__ wbase = W + (size_t)(half ? 2 : 0) * N + (n0 + l);

  v8f acc0 = {}, acc1 = {}, acc2 = {}, acc3 = {};
  #pragma unroll 4
  for (int k = 0; k < K; k += 4) {
    v2f a;
    a.x = xrow[k];
    a.y = xrow[k + 1];
    const float* __restrict__ w0 = wbase + (size_t)k * N;        // row k(+half*2)
    const float* __restrict__ w1 = wbase + (size_t)(k + 1) * N;  // row k+1(+half*2)
    v2f b0; b0.x = w0[0];  b0.y = w1[0];
    v2f b1; b1.x = w0[16]; b1.y = w1[16];
    v2f b2; b2.x = w0[32]; b2.y = w1[32];
    v2f b3; b3.x = w0[48]; b3.y = w1[48];
    acc0 = __builtin_amdgcn_wmma_f32_16x16x4_f32(false, a, false, b0, (short)0,
                                                 acc0, false, false);
    acc1 = __builtin_amdgcn_wmma_f32_16x16x4_f32(false, a, false, b1, (short)0,
                                                 acc1, false, false);
    acc2 = __builtin_amdgcn_wmma_f32_16x16x4_f32(false, a, false, b2, (short)0,
                                                 acc2, false, false);
    acc3 = __builtin_amdgcn_wmma_f32_16x16x4_f32(false, a, false, b3, (short)0,
                                                 acc3, false, false);
  }

  // Epilogue: bias add + store all four 16x16 tiles.
  const int rowBase = m0 + (half ? 8 : 0);
  v8f accs[4] = {acc0, acc1, acc2, acc3};
  #pragma unroll
  for (int t = 0; t < 4; ++t) {
    const int col = n0 + 16 * t + l;
    const float bv = bias[col];
    #pragma unroll
    for (int i = 0; i < 8; ++i) {
      Y[(size_t)(rowBase + i) * N + col] = accs[t][i] + bv;
    }
  }
}

// ---------------------------------------------------------------------------
// Softmax over the 64 (LEVELS*FRAMES*POINTS) logits of each (query, head).
// attn layout: (LQ, HEADS*64) row-major, contiguous 64 per head. In-place.
// ---------------------------------------------------------------------------
__global__ void softmax64(float* __restrict__ attn, int nqh) {
  const int t = blockIdx.x * blockDim.x + threadIdx.x;
  if (t >= nqh) return;
  float* p = attn + (size_t)t * 64;
  float m = -INFINITY;
  #pragma unroll 8
  for (int i = 0; i < 64; ++i) m = fmaxf(m, p[i]);
  float s = 0.f;
  #pragma unroll 8
  for (int i = 0; i < 64; ++i) s += __expf(p[i] - m);
  const float inv = 1.f / s;
  #pragma unroll 8
  for (int i = 0; i < 64; ++i) p[i] = __expf(p[i] - m) * inv;
}

// ---------------------------------------------------------------------------
// loc = reference + temporal_offset + off / norm, in-place over the off buffer.
// off layout per query: [head][level][frame][point][xy]  (1024 floats).
// norm per level: x -> T, y -> H*W (reference's transposed grid convention).
// t indexes (q, head, level, frame, point) pairs: total = LQ*512.
// ---------------------------------------------------------------------------
__global__ void build_loc(float* __restrict__ loc,
                          const float* __restrict__ refp,   // (LQ, LEVELS, 2)
                          const float* __restrict__ toff,   // (LQ, LEVELS, FRAMES, 2)
                          const int* __restrict__ shapes,   // (LEVELS, 3) H,W,T
                          int total) {
  const int t = blockIdx.x * blockDim.x + threadIdx.x;
  if (t >= total) return;
  const int q = t >> 9;               // /512
  const int r = t & 511;
  const int lvl = (r >> 4) & 3;
  const int frame = (r >> 2) & 3;
  const int h = shapes[lvl * 3 + 0];
  const int w = shapes[lvl * 3 + 1];
  const int T = shapes[lvl * 3 + 2];
  const float inv_nx = 1.f / (float)T;
  const float inv_ny = 1.f / (float)(h * w);

  const size_t base = (size_t)t * 2;
  const float ox = loc[base + 0];
  const float oy = loc[base + 1];
  const size_t ri = ((size_t)q * LEVELS_ + lvl) * 2;
  const size_t ti = (((size_t)q * LEVELS_ + lvl) * FRAMES_ + frame) * 2;
  loc[base + 0] = refp[ri + 0] + toff[ti + 0] + ox * inv_nx;
  loc[base + 1] = refp[ri + 1] + toff[ti + 1] + oy * inv_ny;
}

// ---------------------------------------------------------------------------
// Deformable sampling core. One wave32 per (query, head); lane = channel d.
// Grid per level is (hl = H*W rows) x (wl = T cols); idx = y*wl + x.
// value layout: (Len_in, HEADS*DHEAD) -> lane-consecutive loads are coalesced.
// ---------------------------------------------------------------------------
__global__ void ms_sample(const float* __restrict__ value,
                          const float* __restrict__ loc,    // (LQ, HEADS, 64, 2)
                          const float* __restrict__ attnw,  // (LQ, HEADS, 64)
                          const int* __restrict__ shapes,
                          const int* __restrict__ starts,
                          float* __restrict__ tmp,          // (LQ, HEADS*DHEAD)
                          int nqh) {
  const int wave = blockIdx.x * (blockDim.x >> 5) + (threadIdx.x >> 5);
  const int lane = threadIdx.x & 31;
  if (wave >= nqh) return;            // wave-uniform
  const int q = wave >> 3;
  const int head = wave & 7;

  const float* __restrict__ locq = loc + ((size_t)q * HEADS_ + head) * 128;
  const float* __restrict__ aw   = attnw + ((size_t)q * HEADS_ + head) * 64;

  float acc = 0.f;
  for (int lvl = 0; lvl < LEVELS_; ++lvl) {
    const int h = shapes[lvl * 3 + 0];
    const int w = shapes[lvl * 3 + 1];
    const int T = shapes[lvl * 3 + 2];
    const int hl = h * w;             // y-extent (flattened pixels)
    const int wl = T;                 // x-extent (frames)
    const int s0 = starts[lvl];
    const float* __restrict__ vbase =
        value + (size_t)s0 * C_ + head * DHEAD_ + lane;

    for (int fp = 0; fp < FRAMES_ * POINTS_; ++fp) {
      const int j = lvl * 16 + fp;
      const float x = locq[j * 2 + 0] * (float)wl - 0.5f;
      const float y = locq[j * 2 + 1] * (float)hl - 0.5f;
      const float wgt = aw[j];
      const float x0f = floorf(x);
      const float y0f = floorf(y);
      const float fx = x - x0f;
      const float fy = y - y0f;
      const int x0 = (int)x0f;
      const int y0 = (int)y0f;
      const float cw[4] = { (1.f - fx) * (1.f - fy),  // (dy=0,dx=0)
                            fx * (1.f - fy),          // (dy=0,dx=1)
                            (1.f - fx) * fy,          // (dy=1,dx=0)
                            fx * fy };                // (dy=1,dx=1)
      #pragma unroll
      for (int c = 0; c < 4; ++c) {
        const int xi = x0 + (c & 1);
        const int yi = y0 + (c >> 1);
        if (xi >= 0 && xi < wl && yi >= 0 && yi < hl) {   // wave-uniform branch
          acc += wgt * cw[c] * vbase[(size_t)(yi * wl + xi) * C_];
        }
      }
    }
  }
  tmp[((size_t)q * HEADS_ + head) * DHEAD_ + lane] = acc;
}

// ---------------------------------------------------------------------------
extern "C" void kernel_launch(void* const* d_in, const int* in_sizes, int n_in,
                              void* d_out, int out_size, void* d_ws, size_t ws_size,
                              hipStream_t stream) {
  (void)in_sizes; (void)n_in; (void)out_size; (void)ws_size;

  const float* query  = (const float*)d_in[0];   // (1, LQ, C)
  const float* refp   = (const float*)d_in[1];   // (1, LQ, L, 2)
  const float* toff   = (const float*)d_in[2];   // (LQ, L, F, 2)
  const float* inflat = (const float*)d_in[3];   // (1, LQ, C)
  const int*   shapes = (const int*)d_in[4];     // (L, 3)
  const int*   starts = (const int*)d_in[5];     // (L,)
  const float* W_off  = (const float*)d_in[6];   // (C, 1024)
  const float* b_off  = (const float*)d_in[7];
  const float* W_attn = (const float*)d_in[8];   // (C, 512)
  const float* b_attn = (const float*)d_in[9];
  const float* W_val  = (const float*)d_in[10];  // (C, C)
  const float* b_val  = (const float*)d_in[11];
  const float* W_out  = (const float*)d_in[12];  // (C, C)
  const float* b_out  = (const float*)d_in[13];
  float* out = (float*)d_out;

  float* ws = (float*)d_ws;
  float* v_val  = ws;                               // LQ*C
  float* v_attn = v_val  + (size_t)LQ_ * C_;        // LQ*512
  float* v_loc  = v_attn + (size_t)LQ_ * 512;       // LQ*1024 (off -> loc in place)
  float* v_tmp  = v_loc  + (size_t)LQ_ * 1024;      // LQ*C

  const int threads = 256;                          // 8 wave32 per block
  const int wavesPerBlock = threads >> 5;

  auto gemm = [&](const float* X, const float* W, const float* B, float* Y,
                  int M, int N, int K) {
    const int groups = (M >> 4) * (N >> 6);         // 16x64 per wave
    const int blocks = (groups + wavesPerBlock - 1) / wavesPerBlock;
    gemm_f32_wmma<<<blocks, threads, 0, stream>>>(X, W, B, Y, M, N, K);
  };

  // 1) Projections (WMMA fp32)
  gemm(inflat, W_val,  b_val,  v_val,  LQ_, C_,   C_);   // value
  gemm(query,  W_attn, b_attn, v_attn, LQ_, 512,  C_);   // attention logits
  gemm(query,  W_off,  b_off,  v_loc,  LQ_, 1024, C_);   // sampling offsets

  // 2) Softmax over 64 per (q, head)
  {
    const int nqh = LQ_ * HEADS_;
    softmax64<<<(nqh + threads - 1) / threads, threads, 0, stream>>>(v_attn, nqh);
  }

  // 3) Offsets -> sampling locations (in place)
  {
    const int total = LQ_ * 512;  // (q, head, level, frame, point) pairs
    build_loc<<<(total + threads - 1) / threads, threads, 0, stream>>>(
        v_loc, refp, toff, shapes, total);
  }

  // 4) Deformable bilinear sampling (1 wave per (q, head))
  {
    const int nqh = LQ_ * HEADS_;
    const int blocks = (nqh + wavesPerBlock - 1) / wavesPerBlock;
    ms_sample<<<blocks, threads, 0, stream>>>(v_val, v_loc, v_attn, shapes,
                                              starts, v_tmp, nqh);
  }

  // 5) Output projection (WMMA fp32)
  gemm(v_tmp, W_out, b_out, out, LQ_, C_, C_);
}